// CantorAttention_21028159881494
// MI455X (gfx1250) — compile-verified
//
#include <hip/hip_runtime.h>
#include <math.h>

// ---------------------------------------------------------------------------
// CantorAttention for MI455X (gfx1250, wave32)
//   phase 1: qkv = x @ W_qkv + b          (WMMA f32 16x16x4)
//   phase 2: gathered smoothed attention  (VALU, L2-resident gathers)
//   phase 3: out = attn @ W_out + b       (WMMA f32 16x16x4)
// ---------------------------------------------------------------------------

typedef __attribute__((ext_vector_type(2))) float v2f;
typedef __attribute__((ext_vector_type(8))) float v8f;

#define S_LEN    2048
#define D_MODEL  512
#define NHEADS   8
#define HDIM     64
#define KNBR     64
#define QKV_COLS 1536   // [q | k | v] each D_MODEL wide, head-major inside

// ---------------------------------------------------------------------------
// C[M,N] = A[M,K] @ W[K,N] + bias[N], fp32, via V_WMMA_F32_16X16X4_F32.
// One wave computes a 16x64 strip (4 accumulators reuse one A fragment).
// Requires M%16==0, N%64==0, K%4==0 (true for all call sites here).
// ---------------------------------------------------------------------------
__global__ __launch_bounds__(256) void gemm_bias_wmma(
    const float* __restrict__ A, const float* __restrict__ W,
    const float* __restrict__ bias, float* __restrict__ C,
    int M, int N, int K)
{
    const int wave   = blockIdx.x * (blockDim.x >> 5) + (threadIdx.x >> 5);
    const int tilesN = N >> 6;                 // strips of 64 columns
    const int m0     = (wave / tilesN) << 4;
    const int n0     = (wave % tilesN) << 6;
    if (m0 >= M) return;

    const int lane = threadIdx.x & 31;
    const int half = lane >> 4;                // 0: lanes 0-15, 1: lanes 16-31
    const int lr   = lane & 15;

    v8f acc0 = {}; v8f acc1 = {}; v8f acc2 = {}; v8f acc3 = {};

    const float* aRow = A + (size_t)(m0 + lr) * K;

    for (int k0 = 0; k0 < K; k0 += 4) {
        // A fragment (16x4 f32): lane<16 -> K=k0,k0+1 ; lane>=16 -> K=k0+2,k0+3
        v2f a;
        const float* ap = aRow + k0 + 2 * half;
        a.x = ap[0];
        a.y = ap[1];

        // B fragments (4x16 f32): VGPR0 = row k0+2*half, VGPR1 = row k0+2*half+1
        const float* bp = W + (size_t)(k0 + 2 * half) * N + n0 + lr;
        v2f b0, b1, b2, b3;
        b0.x = bp[0];   b0.y = bp[N + 0];
        b1.x = bp[16];  b1.y = bp[N + 16];
        b2.x = bp[32];  b2.y = bp[N + 32];
        b3.x = bp[48];  b3.y = bp[N + 48];

        acc0 = __builtin_amdgcn_wmma_f32_16x16x4_f32(false, a, false, b0,
                                                     (short)0, acc0, false, false);
        acc1 = __builtin_amdgcn_wmma_f32_16x16x4_f32(false, a, false, b1,
                                                     (short)0, acc1, false, false);
        acc2 = __builtin_amdgcn_wmma_f32_16x16x4_f32(false, a, false, b2,
                                                     (short)0, acc2, false, false);
        acc3 = __builtin_amdgcn_wmma_f32_16x16x4_f32(false, a, false, b3,
                                                     (short)0, acc3, false, false);
    }

    // C/D layout: VGPR i -> row m0 + i + 8*half, column n0 + 16*t + lr
    #pragma unroll
    for (int i = 0; i < 8; ++i) {
        const int row = m0 + i + 8 * half;
        float* crow = C + (size_t)row * N + n0 + lr;
        crow[0]  = acc0[i] + bias[n0 + lr];
        crow[16] = acc1[i] + bias[n0 + lr + 16];
        crow[32] = acc2[i] + bias[n0 + lr + 32];
        crow[48] = acc3[i] + bias[n0 + lr + 48];
    }
}

// ---------------------------------------------------------------------------
// Gathered smoothed attention. 1 block = 1 query position, wave h = head h.
//   k_g = 0.5*k[r] + 0.25*(k[r-1] + k[r+1])   (clamped), same for v
//   scores = q . k_g * 1/8 -> softmax -> out = attn . v_g
// qkv row layout: [ q(h*64+d) | k(512 + h*64+d) | v(1024 + h*64+d) ]
// ---------------------------------------------------------------------------
__global__ __launch_bounds__(256) void cantor_attn(
    const float* __restrict__ qkv, const int* __restrict__ routes,
    float* __restrict__ out)
{
    __shared__ int                  r_sm[KNBR];
    __shared__ __align__(16) float  q_sm[NHEADS * HDIM];
    __shared__ __align__(16) float  a_sm[NHEADS * KNBR];

    const int s    = blockIdx.x;
    const int tid  = threadIdx.x;
    const int h    = tid >> 5;      // wave index = head
    const int lane = tid & 31;

    if (tid < KNBR) r_sm[tid] = routes[s * KNBR + tid];
    // stage this query's q row (512 floats) into LDS, 256 threads x float2
    ((float2*)q_sm)[tid] = ((const float2*)(qkv + (size_t)s * QKV_COLS))[tid];
    __syncthreads();

    const float c0 = 0.5f, c1 = 0.25f, scale = 0.125f;
    const float4* q4 = (const float4*)(q_sm + h * HDIM);

    // ---- scores: 2 neighbors per lane, full 64-d dot each -----------------
    float sc[2];
    #pragma unroll
    for (int t = 0; t < 2; ++t) {
        const int j  = lane + 32 * t;
        const int r  = r_sm[j];
        const int rl = max(r - 1, 0);
        const int rr = min(r + 1, S_LEN - 1);
        const float4* kr  = (const float4*)(qkv + (size_t)r  * QKV_COLS + D_MODEL + h * HDIM);
        const float4* krl = (const float4*)(qkv + (size_t)rl * QKV_COLS + D_MODEL + h * HDIM);
        const float4* krr = (const float4*)(qkv + (size_t)rr * QKV_COLS + D_MODEL + h * HDIM);
        float acc = 0.0f;
        #pragma unroll
        for (int u = 0; u < HDIM / 4; ++u) {
            const float4 kc = kr[u], kl = krl[u], kq = krr[u], qv = q4[u];
            acc += qv.x * (c0 * kc.x + c1 * (kl.x + kq.x));
            acc += qv.y * (c0 * kc.y + c1 * (kl.y + kq.y));
            acc += qv.z * (c0 * kc.z + c1 * (kl.z + kq.z));
            acc += qv.w * (c0 * kc.w + c1 * (kl.w + kq.w));
        }
        sc[t] = acc * scale;
    }

    // ---- wave32 softmax over 64 scores ------------------------------------
    float m = fmaxf(sc[0], sc[1]);
    #pragma unroll
    for (int off = 16; off > 0; off >>= 1) m = fmaxf(m, __shfl_xor(m, off, 32));
    const float e0 = __expf(sc[0] - m);
    const float e1 = __expf(sc[1] - m);
    float sum = e0 + e1;
    #pragma unroll
    for (int off = 16; off > 0; off >>= 1) sum += __shfl_xor(sum, off, 32);
    const float inv = 1.0f / sum;
    a_sm[h * KNBR + lane]      = e0 * inv;
    a_sm[h * KNBR + lane + 32] = e1 * inv;
    __syncthreads();

    // ---- out = attn . v_g ; lane owns 2 dims -> fully coalesced row reads --
    float ox = 0.0f, oy = 0.0f;
    #pragma unroll 4
    for (int j = 0; j < KNBR; ++j) {
        const float w  = a_sm[h * KNBR + j];
        const int   r  = r_sm[j];
        const int   rl = max(r - 1, 0);
        const int   rr = min(r + 1, S_LEN - 1);
        const float2 vc = ((const float2*)(qkv + (size_t)r  * QKV_COLS + 2 * D_MODEL + h * HDIM))[lane];
        const float2 vl = ((const float2*)(qkv + (size_t)rl * QKV_COLS + 2 * D_MODEL + h * HDIM))[lane];
        const float2 vq = ((const float2*)(qkv + (size_t)rr * QKV_COLS + 2 * D_MODEL + h * HDIM))[lane];
        ox += w * (c0 * vc.x + c1 * (vl.x + vq.x));
        oy += w * (c0 * vc.y + c1 * (vl.y + vq.y));
    }
    float2* orow = (float2*)(out + (size_t)s * D_MODEL + h * HDIM);
    orow[lane] = make_float2(ox, oy);
}

// ---------------------------------------------------------------------------
extern "C" void kernel_launch(void* const* d_in, const int* in_sizes, int n_in,
                              void* d_out, int out_size, void* d_ws, size_t ws_size,
                              hipStream_t stream)
{
    const float* x      = (const float*)d_in[0];
    const int*   routes = (const int*)  d_in[1];
    const float* W_qkv  = (const float*)d_in[2];
    const float* b_qkv  = (const float*)d_in[3];
    const float* W_out  = (const float*)d_in[4];
    const float* b_out  = (const float*)d_in[5];
    float*       outp   = (float*)d_out;

    float* ws       = (float*)d_ws;
    float* qkv_ws   = ws;                                   // 2048 x 1536
    float* attn_ws  = ws + (size_t)S_LEN * QKV_COLS;        // 2048 x 512

    // GEMM1: qkv = x @ W_qkv + b_qkv   (2048x512 x 512x1536)
    {
        const int waves  = (S_LEN / 16) * (QKV_COLS / 64);  // 3072
        gemm_bias_wmma<<<waves / 8, 256, 0, stream>>>(
            x, W_qkv, b_qkv, qkv_ws, S_LEN, QKV_COLS, D_MODEL);
    }

    // Attention: one block per query, one wave per head
    cantor_attn<<<S_LEN, 256, 0, stream>>>(qkv_ws, routes, attn_ws);

    // GEMM2: out = attn @ W_out + b_out  (2048x512 x 512x512)
    {
        const int waves  = (S_LEN / 16) * (D_MODEL / 64);   // 1024
        gemm_bias_wmma<<<waves / 8, 256, 0, stream>>>(
            attn_ws, W_out, b_out, outp, S_LEN, D_MODEL, D_MODEL);
    }
}